// TwoChannelTraceLIF_MinWTA_64450279244428
// MI455X (gfx1250) — compile-verified
//
#include <hip/hip_runtime.h>

typedef __attribute__((ext_vector_type(2))) float v2f;
typedef __attribute__((ext_vector_type(8))) float v8f;

#define B_    64
#define L_    8192
#define WARM  8
#define LP    (L_ - WARM)              /* 8184 */
#define KCH   32
#define NELEM ((size_t)B_ * LP * KCH)  /* 16,760,832 per tensor */
#define CH_STRIDE (6 * NELEM + (size_t)B_ + (size_t)(B_ * KCH) + 1)
#define NTILES ((LP + 15) / 16)        /* 512 */

// ---------------------------------------------------------------------------
// Kernel 1: filter bank as WMMA GEMM. One wave handles a 16-timestep x 32-ch
// tile for one (ch, b). Per tile: 2 WMMAs (k=5 filter, taps padded to 8) +
// 3 WMMAs (k=9 filter, taps padded to 12), f32 16x16x4.
// ---------------------------------------------------------------------------
__global__ __launch_bounds__(256)
void conv_wmma_kernel(const float* __restrict__ x,  const float* __restrict__ y,
                      const float* __restrict__ wx5, const float* __restrict__ bx5,
                      const float* __restrict__ wx9, const float* __restrict__ bx9,
                      const float* __restrict__ wy5, const float* __restrict__ by5,
                      const float* __restrict__ wy9, const float* __restrict__ by9,
                      float* __restrict__ out)
{
    const int lane   = threadIdx.x & 31;
    const int waveId = threadIdx.x >> 5;
    int gw   = blockIdx.x * 8 + waveId;          // global wave id
    int tile = gw % NTILES;
    int rem  = gw / NTILES;
    int b    = rem % B_;
    int ch   = rem / B_;

    const float* xs = (ch == 0 ? x : y) + (size_t)b * L_;
    float* d2 = out + (size_t)ch * CH_STRIDE;    // d2 is first tensor in channel

    const int i0   = tile * 16;
    const int m    = lane & 15;                  // A-matrix row (M) for this lane
    const int koff = (lane >> 4) * 2;            // A layout: lanes 0-15 -> K0,K1; 16-31 -> K2,K3
    const int c    = lane & 15;                  // B/D column (N)

    for (int g = 0; g < 2; ++g) {
        const int    ktap   = (g == 0) ? 5 : 9;
        const int    nchunk = (g == 0) ? 2 : 3;  // taps padded to multiples of 4
        const float* wp = (g == 0) ? (ch == 0 ? wx5 : wy5) : (ch == 0 ? wx9 : wy9);
        const float* bp = (g == 0) ? (ch == 0 ? bx5 : by5) : (ch == 0 ? bx9 : by9);
        const int chbase = g * 16;

        v8f acc = {0.f, 0.f, 0.f, 0.f, 0.f, 0.f, 0.f, 0.f};
        for (int kc = 0; kc < nchunk; ++kc) {
            const int j0 = kc * 4 + koff;
            v2f a, bm;
            // A: shifted input window, y[c,t] = sum_j w[c,j] * x[t-(ktap-1)+j]
            {
                int j = j0;
                int t = i0 + WARM + m - (ktap - 1) + j;
                t = (t > L_ - 1) ? (L_ - 1) : t;               // clamp for partial last tile
                a.x = (j < ktap) ? xs[t] : 0.f;
                j = j0 + 1;
                t = i0 + WARM + m - (ktap - 1) + j;
                t = (t > L_ - 1) ? (L_ - 1) : t;
                a.y = (j < ktap) ? xs[t] : 0.f;
            }
            // B: filter taps, [C,1,ktap] row-major
            {
                int j = j0;
                bm.x = (j < ktap) ? wp[c * ktap + j] : 0.f;
                j = j0 + 1;
                bm.y = (j < ktap) ? wp[c * ktap + j] : 0.f;
            }
            acc = __builtin_amdgcn_wmma_f32_16x16x4_f32(
                false, a, false, bm, (short)0, acc, false, false);
        }

        // D layout: VGPR r -> M = r (lanes 0-15) / r+8 (lanes 16-31), N = lane&15
        const float bias  = bp[c];
        const int   mbase = (lane >> 4) * 8;
        for (int r = 0; r < 8; ++r) {
            int i = i0 + mbase + r;
            if (i < LP) {
                float v = acc[r] + bias;
                d2[((size_t)b * LP + i) * KCH + chbase + c] = (v > 0.f) ? v : 0.f;
            }
        }
    }
}

// ---------------------------------------------------------------------------
// Kernel 2: fused trace (EMA) + adaptive-LIF scan. One thread per (ch,b,k);
// lane = k so all global accesses are 128B-coalesced per wave32.
// Also records first-spike time & total spikes per sequence for the WTA.
// ---------------------------------------------------------------------------
__global__ __launch_bounds__(256)
void scan_kernel(float* __restrict__ out,
                 int* __restrict__ tfirst_ws, float* __restrict__ ssum_ws)
{
    int tid = blockIdx.x * blockDim.x + threadIdx.x;
    if (tid >= 2 * B_ * KCH) return;
    const int k  = tid & 31;
    const int b  = (tid >> 5) & 63;
    const int ch = tid >> 11;

    float* chb = out + (size_t)ch * CH_STRIDE;
    const float* d2   = chb;
    float* z_out  = chb + 1 * NELEM;
    float* s_out  = chb + 2 * NELEM;
    float* ss_out = chb + 3 * NELEM;
    float* v_out  = chb + 4 * NELEM;

    const size_t base = ((size_t)b * LP) * KCH + (size_t)k;
    float z = 0.f, v = 0.f, a = 0.f, snum = 0.f;
    int tf = LP;

    for (int i = 0; i < LP; ++i) {
        size_t e = base + (size_t)i * KCH;
        __builtin_prefetch(d2 + e + (size_t)96 * KCH, 0, 1);  // global_prefetch_b8
        float d = d2[e];
        // trace: z = lam*z + (1-lam)*d
        z = 0.95f * z + 0.05f * d;
        z_out[e] = z;
        // LIF (forward values only; STE makes s == s_hard numerically)
        v = v - v * (1.0f / 20.0f) + z;
        float th   = 1.5f + 1.5f * a;
        float vpre = v;
        float s    = (v >= th) ? 1.f : 0.f;
        snum += s;
        v = (s > 0.f) ? -0.5f : v;           // reset
        a = a - a * (1.0f / 100.0f) + s;     // adaptation
        s_out[e]  = s;
        ss_out[e] = snum;
        v_out[e]  = vpre;
        if (s > 0.5f && tf == LP) tf = i;
    }
    tfirst_ws[tid] = tf;
    ssum_ws[tid]   = snum;
}

// ---------------------------------------------------------------------------
// Kernel 3: WTA reduction. One wave32 block per (ch,b). Forward value of w is
// exactly one_hot(idx) due to the straight-through estimator.
// ---------------------------------------------------------------------------
__global__ __launch_bounds__(32)
void wta_kernel(float* __restrict__ out,
                const int* __restrict__ tfirst_ws, const float* __restrict__ ssum_ws)
{
    __shared__ int   stf[KCH];
    __shared__ float ssm[KCH];
    __shared__ int   sidx;
    const int ch = blockIdx.x >> 6;
    const int b  = blockIdx.x & 63;
    const int k  = threadIdx.x;
    const int base = ch * B_ * KCH + b * KCH;
    stf[k] = tfirst_ws[base + k];
    ssm[k] = ssum_ws[base + k];
    __syncthreads();
    if (k == 0) {
        int tany = LP;
        for (int j = 0; j < KCH; ++j) tany = (stf[j] < tany) ? stf[j] : tany;
        int kstar = 0;
        for (int j = KCH - 1; j >= 0; --j) if (stf[j] == tany) kstar = j; // first k spiking at t*
        int kfb = 0; float best = ssm[0];
        for (int j = 1; j < KCH; ++j) if (ssm[j] > best) { best = ssm[j]; kfb = j; }
        int idx = (tany < LP) ? kstar : kfb;
        sidx = idx;
        float* chb = out + (size_t)ch * CH_STRIDE;
        chb[5 * NELEM + b] = (float)idx;                         // idx output
        if (b == 0) chb[6 * NELEM + B_ + B_ * KCH] = (float)WARM; // trailing WARMUP scalar
    }
    __syncthreads();
    float* chb = out + (size_t)ch * CH_STRIDE;
    chb[5 * NELEM + B_ + (size_t)b * KCH + k] = (k == sidx) ? 1.f : 0.f; // w (one-hot)
}

// ---------------------------------------------------------------------------
// Kernel 4: s_gated = s * w[b,k] (w is one-hot). s is hot in the 192MB L2.
// ---------------------------------------------------------------------------
__global__ __launch_bounds__(256)
void gate_kernel(float* __restrict__ out)
{
    size_t e = (size_t)blockIdx.x * blockDim.x + threadIdx.x;
    if (e >= 2 * NELEM) return;
    const int    ch = (int)(e / NELEM);
    const size_t r  = e % NELEM;
    float* chb = out + (size_t)ch * CH_STRIDE;
    const int k = (int)(r & 31);
    const int b = (int)(r / ((size_t)LP * KCH));
    float w = chb[5 * NELEM + B_ + (size_t)b * KCH + k];
    chb[5 * NELEM + B_ + B_ * KCH + r] = chb[2 * NELEM + r] * w;
}

// ---------------------------------------------------------------------------
extern "C" void kernel_launch(void* const* d_in, const int* in_sizes, int n_in,
                              void* d_out, int out_size, void* d_ws, size_t ws_size,
                              hipStream_t stream)
{
    const float* x   = (const float*)d_in[0];
    const float* y   = (const float*)d_in[1];
    const float* wx5 = (const float*)d_in[2];
    const float* bx5 = (const float*)d_in[3];
    const float* wx9 = (const float*)d_in[4];
    const float* bx9 = (const float*)d_in[5];
    const float* wy5 = (const float*)d_in[6];
    const float* by5 = (const float*)d_in[7];
    const float* wy9 = (const float*)d_in[8];
    const float* by9 = (const float*)d_in[9];
    float* out = (float*)d_out;

    int*   tf_ws = (int*)d_ws;
    float* ss_ws = (float*)((char*)d_ws + 2 * B_ * KCH * sizeof(int));

    // 1) filter bank via f32 WMMA: 2*64*512 waves, 8 waves (256 thr) / block
    conv_wmma_kernel<<<(2 * B_ * NTILES) / 8, 256, 0, stream>>>(
        x, y, wx5, bx5, wx9, bx9, wy5, by5, wy9, by9, out);

    // 2) fused trace + LIF sequential scan, 4096 threads
    scan_kernel<<<(2 * B_ * KCH + 255) / 256, 256, 0, stream>>>(out, tf_ws, ss_ws);

    // 3) WTA: one wave per (ch, b)
    wta_kernel<<<2 * B_, 32, 0, stream>>>(out, tf_ws, ss_ws);

    // 4) gate spikes by one-hot winner
    size_t total = 2 * NELEM;
    gate_kernel<<<(unsigned)((total + 255) / 256), 256, 0, stream>>>(out);
}